// SMPLA_parser_55576876810725
// MI455X (gfx1250) — compile-verified
//
#include <hip/hip_runtime.h>
#include <math.h>

// ---------------- problem constants ----------------
#define NB    1024
#define NV    6890
#define NJ    24
#define NBETA 10
#define NPC   (3*NV)        // 20670 columns of the big GEMM
#define KT    220           // 207 pose feats + 10 betas + 1 (template) + 2 pad
#define PSTR  120           // LDS row stride for P tile: 96 data + TDM pad (8 DW per 32 DW)

// workspace layout (floats)
#define OFF_JRT 0           // 72
#define OFF_JRS 128         // 720
#define OFF_A   1024        // 1024*24*12 = 294912
#define OFF_F   295936      // 1024*220  = 225280
#define OFF_P   521216      // 220*20670 = 4547400

typedef __attribute__((ext_vector_type(2))) float v2f;
typedef __attribute__((ext_vector_type(8))) float v8f;
typedef unsigned int u32x4 __attribute__((ext_vector_type(4)));
typedef int          i32x4 __attribute__((ext_vector_type(4)));
typedef int          i32x8 __attribute__((ext_vector_type(8)));

// ---------------------------------------------------------------------------
// Kernel 0: build fused RHS matrix P (KT x NPC):
//   rows 0..206  : posedirs ; rows 207..216: shapedirs transposed
//   row  217     : v_template ; rows 218,219: zero pad (KT % 4 == 0)
// ---------------------------------------------------------------------------
__global__ void build_P(const float* __restrict__ posedirs,
                        const float* __restrict__ shapedirs,
                        const float* __restrict__ v_template,
                        float* __restrict__ P) {
  int idx = blockIdx.x * 256 + threadIdx.x;
  if (idx >= KT * NPC) return;
  int r = idx / NPC, c = idx % NPC;
  float val;
  if (r < 207) {
    val = posedirs[(size_t)r * NPC + c];
  } else if (r < 217) {
    int l = r - 207;
    int v = c / 3, k = c % 3;
    val = shapedirs[((size_t)v * 3 + k) * NBETA + l];
  } else if (r == 217) {
    val = v_template[c];
  } else {
    val = 0.0f;
  }
  P[idx] = val;
}

// ---------------------------------------------------------------------------
// Kernel 1: fold J_regressor once:
//   JRT[j,k] = sum_v Jreg[j,v]*v_template[v,k] ; JRS[j,k,l] likewise w/ shapedirs
// ---------------------------------------------------------------------------
__global__ void jreg_reduce(const float* __restrict__ Jreg,
                            const float* __restrict__ v_template,
                            const float* __restrict__ shapedirs,
                            float* __restrict__ JRT,
                            float* __restrict__ JRS) {
  int j   = blockIdx.x / 33;
  int col = blockIdx.x % 33;
  __shared__ float red[256];
  float s = 0.0f;
  for (int v = threadIdx.x; v < NV; v += 256) {
    float jr = Jreg[(size_t)j * NV + v];
    float x;
    if (col < 3) {
      x = v_template[v * 3 + col];
    } else {
      int k = (col - 3) / NBETA, l = (col - 3) % NBETA;
      x = shapedirs[((size_t)v * 3 + k) * NBETA + l];
    }
    s += jr * x;
  }
  red[threadIdx.x] = s;
  __syncthreads();
  for (int w = 128; w > 0; w >>= 1) {
    if (threadIdx.x < w) red[threadIdx.x] += red[threadIdx.x + w];
    __syncthreads();
  }
  if (threadIdx.x == 0) {
    if (col < 3) JRT[j * 3 + col] = red[0];
    else {
      int k = (col - 3) / NBETA, l = (col - 3) % NBETA;
      JRS[(j * 3 + k) * NBETA + l] = red[0];
    }
  }
}

// ---------------------------------------------------------------------------
// Kernel 2: per-batch prep (Rodrigues, pose-feature row F, kinematic chain, A)
// ---------------------------------------------------------------------------
__global__ void prep_batch(const float* __restrict__ betas,
                           const float* __restrict__ pose,
                           const float* __restrict__ JRT,
                           const float* __restrict__ JRS,
                           float* __restrict__ Aout,
                           float* __restrict__ Fout) {
  int b = blockIdx.x * blockDim.x + threadIdx.x;
  if (b >= NB) return;
  const int par[NJ] = {0,0,0,0,1,2,3,4,5,6,7,8,9,9,9,12,13,14,16,17,18,19,20,21};

  float bet[NBETA];
  for (int l = 0; l < NBETA; ++l) bet[l] = betas[b * NBETA + l];

  float Jl[NJ][3];
  for (int j = 0; j < NJ; ++j)
    for (int k = 0; k < 3; ++k) {
      float s = JRT[j * 3 + k];
      for (int l = 0; l < NBETA; ++l) s += JRS[(j * 3 + k) * NBETA + l] * bet[l];
      Jl[j][k] = s;
    }

  float R[NJ][9];
  for (int j = 0; j < NJ; ++j) {
    float x = pose[b * 72 + 3 * j + 0];
    float y = pose[b * 72 + 3 * j + 1];
    float z = pose[b * 72 + 3 * j + 2];
    float xp = x + 1e-8f, yp = y + 1e-8f, zp = z + 1e-8f;
    float ang = sqrtf(xp * xp + yp * yp + zp * zp);
    float inv = 1.0f / ang;
    float rx = x * inv, ry = y * inv, rz = z * inv;
    float c = cosf(ang), s = sinf(ang), t = 1.0f - c;
    R[j][0] = 1.0f - t * (ry * ry + rz * rz);
    R[j][1] = -s * rz + t * rx * ry;
    R[j][2] =  s * ry + t * rx * rz;
    R[j][3] =  s * rz + t * rx * ry;
    R[j][4] = 1.0f - t * (rx * rx + rz * rz);
    R[j][5] = -s * rx + t * ry * rz;
    R[j][6] = -s * ry + t * rx * rz;
    R[j][7] =  s * rx + t * ry * rz;
    R[j][8] = 1.0f - t * (rx * rx + ry * ry);
  }

  float* F = Fout + (size_t)b * KT;
  for (int j = 1; j < NJ; ++j)
    for (int e = 0; e < 9; ++e)
      F[(j - 1) * 9 + e] = R[j][e] - ((e == 0 || e == 4 || e == 8) ? 1.0f : 0.0f);
  for (int l = 0; l < NBETA; ++l) F[207 + l] = bet[l];
  F[217] = 1.0f; F[218] = 0.0f; F[219] = 0.0f;

  float G[NJ][12];
  for (int j = 0; j < NJ; ++j) {
    float rel[3];
    for (int k = 0; k < 3; ++k) rel[k] = Jl[j][k] - (j ? Jl[par[j]][k] : 0.0f);
    if (j == 0) {
      for (int p = 0; p < 3; ++p) {
        G[0][p * 4 + 0] = R[0][p * 3 + 0];
        G[0][p * 4 + 1] = R[0][p * 3 + 1];
        G[0][p * 4 + 2] = R[0][p * 3 + 2];
        G[0][p * 4 + 3] = rel[p];
      }
    } else {
      int pp = par[j];
      for (int p = 0; p < 3; ++p) {
        for (int q = 0; q < 3; ++q)
          G[j][p * 4 + q] = G[pp][p * 4 + 0] * R[j][0 * 3 + q] +
                            G[pp][p * 4 + 1] * R[j][1 * 3 + q] +
                            G[pp][p * 4 + 2] * R[j][2 * 3 + q];
        G[j][p * 4 + 3] = G[pp][p * 4 + 0] * rel[0] +
                          G[pp][p * 4 + 1] * rel[1] +
                          G[pp][p * 4 + 2] * rel[2] + G[pp][p * 4 + 3];
      }
    }
  }

  float* Ab = Aout + (size_t)b * (NJ * 12);
  for (int j = 0; j < NJ; ++j)
    for (int p = 0; p < 3; ++p) {
      float r0 = G[j][p * 4 + 0], r1 = G[j][p * 4 + 1], r2 = G[j][p * 4 + 2];
      Ab[j * 12 + p * 4 + 0] = r0;
      Ab[j * 12 + p * 4 + 1] = r1;
      Ab[j * 12 + p * 4 + 2] = r2;
      Ab[j * 12 + p * 4 + 3] = G[j][p * 4 + 3] -
          (r0 * Jl[j][0] + r1 * Jl[j][1] + r2 * Jl[j][2]);
    }
}

// ---------------------------------------------------------------------------
// TDM helper: issue a 2D tensor_load_to_lds (D# per ISA 8.3/8.4).
//   tile_dim0 x tile_dim1 elements of 4 bytes; row stride td0_stride elems;
//   tensor_dim0 used for OOB zero-fill in X; optional LDS pad 8DW per 32DW.
// ---------------------------------------------------------------------------
__device__ __forceinline__
void tdm_load_2d(unsigned lds_addr, const float* gptr,
                 unsigned tensor_dim0, unsigned tensor_dim1,
                 unsigned tile_dim0, unsigned tile_dim1,
                 unsigned td0_stride, bool pad) {
#if __has_builtin(__builtin_amdgcn_tensor_load_to_lds)
  unsigned long long ga = (unsigned long long)(uintptr_t)gptr;
  u32x4 g0;
  g0[0] = 1u;                                   // count=1, user mode
  g0[1] = lds_addr;                             // LDS byte address
  g0[2] = (unsigned)(ga & 0xFFFFFFFFu);         // global_addr lo
  g0[3] = (unsigned)((ga >> 32) & 0x01FFFFFFu)  // global_addr hi (57-bit)
          | 0x80000000u;                        // type = 2 ("image")
  i32x8 g1;
  unsigned dw0 = (2u << 16);                    // data_size = 4B
  if (pad) dw0 |= (1u << 20)                    // pad_enable
             |  (4u << 22)                      // pad_interval: 32 DWORDs
             |  (7u << 25);                     // pad_amount:   8 DWORDs
  g1[0] = (int)dw0;
  g1[1] = (int)((tensor_dim0 & 0xFFFFu) << 16);             // td0[15:0]
  g1[2] = (int)(((tensor_dim0 >> 16) & 0xFFFFu)             // td0[31:16]
                | ((tensor_dim1 & 0xFFFFu) << 16));         // td1[15:0]
  g1[3] = (int)(((tensor_dim1 >> 16) & 0xFFFFu)
                | ((tile_dim0 & 0xFFFFu) << 16));           // tile_dim0
  g1[4] = (int)(tile_dim1 & 0xFFFFu);                       // tile_dim1, tile_dim2=0
  g1[5] = (int)td0_stride;                                  // dim0 stride lo
  g1[6] = 0;                                                // stride hi, dim1 stride
  g1[7] = 0;
  i32x4 z4 = {0, 0, 0, 0};
#if __clang_major__ >= 23
  i32x8 z8 = {0, 0, 0, 0, 0, 0, 0, 0};
  __builtin_amdgcn_tensor_load_to_lds(g0, g1, z4, z4, z8, 0);
#else
  __builtin_amdgcn_tensor_load_to_lds(g0, g1, z4, z4, 0);
#endif
#else
  (void)lds_addr; (void)gptr; (void)tensor_dim0; (void)tensor_dim1;
  (void)tile_dim0; (void)tile_dim1; (void)td0_stride; (void)pad;
#endif
}

// ---------------------------------------------------------------------------
// Kernel 3 (main, fused): WG = 16 batches x 32 vertices (96 GEMM columns).
// Stage: TDM loads P tile (220x96, padded stride 120) + F tile (16x220) to LDS.
// Phase A: WMMA f32 16x16x4 GEMM  v_posed = F(16xKT) @ P(KTx96), 6 waves.
// Phase B: LBS blend Tv = sum_j w[v,j]*A[b,j]; verts = Tv * [v_posed,1].
// ---------------------------------------------------------------------------
__global__ __launch_bounds__(192)
void lbs_main(const float* __restrict__ Fws,
              const float* __restrict__ Pws,
              const float* __restrict__ Aws,
              const float* __restrict__ lbsw,
              float* __restrict__ out) {
  extern __shared__ float smem[];
  float* sP = smem;                    // KT * PSTR    = 26400
  float* sF = sP + KT * PSTR;          // 16 * KT      = 3520  (row-major [b][k])
  float* sC = sF + 16 * KT;            // 16 * 100     = 1600
  float* sA = sC + 16 * 100;           // 16 * 24 * 12 = 4608
  float* sW = sA + 16 * 288;           // 32 * 25      = 800
                                       // total 36928 floats = 147,712 B

  const int tid  = threadIdx.x;
  const int lane = tid & 31;
  const int wave = tid >> 5;           // 0..5
  const int v0   = blockIdx.x * 32;    // first vertex of tile
  const int c0   = blockIdx.x * 96;    // first GEMM column
  const int b0   = blockIdx.y * 16;    // first batch of tile

#if __has_builtin(__builtin_amdgcn_tensor_load_to_lds)
  if (wave == 0) {                     // TDM ignores EXEC: must branch-gate
    unsigned ldsP = (unsigned)(unsigned long long)(uintptr_t)(void*)sP;
    unsigned ldsF = (unsigned)(unsigned long long)(uintptr_t)(void*)sF;
    // P tile: 96 cols x 220 rows; tensor_dim0 = cols remaining -> OOB zero-fill
    tdm_load_2d(ldsP, Pws + (size_t)c0, (unsigned)(NPC - c0), KT,
                96u, KT, (unsigned)NPC, /*pad=*/true);
    // F tile: 220 cols x 16 rows, contiguous
    tdm_load_2d(ldsF, Fws + (size_t)b0 * KT, KT, 16u, KT, 16u, KT, /*pad=*/false);
    __builtin_amdgcn_s_wait_tensorcnt(0);
  }
#else
  // fallback: plain cooperative loads into identical layouts
  for (int i = tid; i < KT * 96; i += 192) {
    int r = i / 96, cc = i % 96;
    int gc = c0 + cc;
    sP[r * PSTR + cc + 8 * (cc / 32)] = (gc < NPC) ? Pws[(size_t)r * NPC + gc] : 0.0f;
  }
  for (int i = tid; i < 16 * KT; i += 192)
    sF[i] = Fws[(size_t)b0 * KT + i];
#endif

  // small tiles: plain loads (A is contiguous; W needs a bounds guard)
  for (int i = tid; i < 16 * 288; i += 192)
    sA[i] = Aws[(size_t)b0 * 288 + i];
  for (int i = tid; i < 32 * 24; i += 192) {
    int vl = i / 24, j = i % 24;
    int v = v0 + vl;
    sW[vl * 25 + j] = (v < NV) ? lbsw[(size_t)v * 24 + j] : 0.0f;
  }
  __syncthreads();

  // ---- Phase A: WMMA GEMM, K loop in steps of 4 ----
  const int half = (lane >> 4) & 1;    // ISA f32 A/B layout: half 0 -> K,K+1 ; 1 -> K+2,K+3
  const int m16  = lane & 15;
  const int nw   = 16 * wave + m16;                 // GEMM column within tile
  const int noff = nw + 8 * (wave >> 1);            // padded LDS column offset
  const int arow = m16 * KT;                        // A row base in sF
  v8f acc = {0.f, 0.f, 0.f, 0.f, 0.f, 0.f, 0.f, 0.f};

  for (int k = 0; k < KT; k += 4) {
    int k2 = k + 2 * half;
    v2f a, bmat;
    a.x    = sF[arow + k2];                         // consecutive dwords
    a.y    = sF[arow + k2 + 1];
    bmat.x = sP[k2 * PSTR + noff];
    bmat.y = sP[(k2 + 1) * PSTR + noff];
    acc = __builtin_amdgcn_wmma_f32_16x16x4_f32(
        false, a, false, bmat, (short)0, acc, false, false);
  }

  // spill C tile to LDS so Phase B can regroup (x,y,z of a vertex together)
#pragma unroll
  for (int r = 0; r < 8; ++r) {
    int mm = r + 8 * half;             // C/D layout: VGPR r -> M = r + 8*half
    sC[mm * 100 + nw] = acc[r];
  }
  __syncthreads();

  // ---- Phase B: LBS blend + final affine transform ----
  for (int p = tid; p < 16 * 32; p += 192) {
    int mm = p >> 5;                   // batch within tile
    int vl = p & 31;                   // vertex within tile
    int v = v0 + vl;
    if (v >= NV) continue;
    float x = sC[mm * 100 + 3 * vl + 0];
    float y = sC[mm * 100 + 3 * vl + 1];
    float z = sC[mm * 100 + 3 * vl + 2];

    float T[12];
#pragma unroll
    for (int q = 0; q < 12; ++q) T[q] = 0.0f;
    const float* Ab = &sA[mm * 288];
    const float* wr = &sW[vl * 25];
    for (int j = 0; j < NJ; ++j) {
      float wj = wr[j];
      const float* Aj = Ab + j * 12;
#pragma unroll
      for (int q = 0; q < 12; ++q) T[q] += wj * Aj[q];
    }
    float ox = T[0] * x + T[1] * y + T[2]  * z + T[3];
    float oy = T[4] * x + T[5] * y + T[6]  * z + T[7];
    float oz = T[8] * x + T[9] * y + T[10] * z + T[11];

    size_t o = ((size_t)(b0 + mm) * NV + v) * 3;
    out[o + 0] = ox;
    out[o + 1] = oy;
    out[o + 2] = oz;
  }
}

// ---------------------------------------------------------------------------
extern "C" void kernel_launch(void* const* d_in, const int* in_sizes, int n_in,
                              void* d_out, int out_size, void* d_ws, size_t ws_size,
                              hipStream_t stream) {
  (void)in_sizes; (void)n_in; (void)out_size; (void)ws_size;
  const float* betas      = (const float*)d_in[0];
  const float* pose       = (const float*)d_in[1];
  const float* v_template = (const float*)d_in[2];
  const float* shapedirs  = (const float*)d_in[3];
  const float* posedirs   = (const float*)d_in[4];
  const float* Jreg       = (const float*)d_in[5];
  const float* lbsw       = (const float*)d_in[6];
  float* out = (float*)d_out;

  float* ws  = (float*)d_ws;
  float* JRT = ws + OFF_JRT;
  float* JRS = ws + OFF_JRS;
  float* Aws = ws + OFF_A;
  float* Fws = ws + OFF_F;
  float* Pws = ws + OFF_P;

  int nP = KT * NPC;
  hipLaunchKernelGGL(build_P, dim3((nP + 255) / 256), dim3(256), 0, stream,
                     posedirs, shapedirs, v_template, Pws);
  hipLaunchKernelGGL(jreg_reduce, dim3(24 * 33), dim3(256), 0, stream,
                     Jreg, v_template, shapedirs, JRT, JRS);
  hipLaunchKernelGGL(prep_batch, dim3(16), dim3(64), 0, stream,
                     betas, pose, JRT, JRS, Aws, Fws);
  size_t smem_bytes = (size_t)(KT * PSTR + 16 * KT + 16 * 100 + 16 * 288 + 32 * 25)
                      * sizeof(float);   // 147,712 B
  hipLaunchKernelGGL(lbs_main, dim3((NV + 31) / 32, NB / 16), dim3(192),
                     smem_bytes, stream, Fws, Pws, Aws, lbsw, out);
}